// HybridNN_76768245449488
// MI455X (gfx1250) — compile-verified
//
#include <hip/hip_runtime.h>
#include <math.h>

typedef __attribute__((ext_vector_type(2))) float v2f;
typedef __attribute__((ext_vector_type(8))) float v8f;

#define BLOCK 256
#define QLAYERS 4

struct cplx { float x, y; };
__device__ __forceinline__ cplx cmul(cplx a, cplx b) {
    return { a.x*b.x - a.y*b.y, a.x*b.y + a.y*b.x };
}
// acc + a*b (complex FMA)
__device__ __forceinline__ cplx cmac(cplx acc, cplx a, cplx b) {
    acc.x = fmaf(a.x, b.x, fmaf(-a.y, b.y, acc.x));
    acc.y = fmaf(a.x, b.y, fmaf( a.y, b.x, acc.y));
    return acc;
}
// tanh: prefer CDNA5 hardware v_tanh_f32; fall back to branchless exp form.
__device__ __forceinline__ float tanh_fast(float v) {
#if __has_builtin(__builtin_amdgcn_tanhf)
    return __builtin_amdgcn_tanhf(v);
#elif __has_builtin(__builtin_amdgcn_tanh_f32)
    return __builtin_amdgcn_tanh_f32(v);
#else
    float t = __expf(-2.0f * fabsf(v));
    float r = __fdividef(1.0f - t, 1.0f + t);
    return copysignf(r, v);
#endif
}

__global__ __launch_bounds__(BLOCK) void hybrid_qnn_kernel(
    const float* __restrict__ x,
    const float* __restrict__ w1, const float* __restrict__ b1,
    const float* __restrict__ w2, const float* __restrict__ b2,
    const float* __restrict__ qw,
    const float* __restrict__ pw, const float* __restrict__ pb,
    float* __restrict__ out, int B)
{
    __shared__ float xs [BLOCK*2];        // staged inputs
    __shared__ float h1s[BLOCK*4];        // layer-1 activations, row-major per sample
    __shared__ float h2s[BLOCK*2];        // layer-2 pre-activations
    __shared__ float gbuf[2*QLAYERS][8];  // Rot gates: [layer*2+wire][Rij re/im]

    const int t    = threadIdx.x;
    const int lane = t & 31;
    const int wave = t >> 5;
    const int hi   = lane >> 4;      // which 16-lane half
    const int N    = lane & 15;      // B/C/D column; also A row M
    const int g0   = blockIdx.x * BLOCK + t;
    const int gc   = (g0 < B) ? g0 : (B - 1);   // clamp: full blocks keep EXEC==all-ones

    xs[2*t]   = x[2*gc];
    xs[2*t+1] = x[2*gc+1];

    // Precompute Rot(phi,theta,omega) 2x2 complex gates once per block (8 threads).
    if (t < 2*QLAYERS) {
        const float phi = qw[t*3+0], th = qw[t*3+1], om = qw[t*3+2];
        float st, ct;  sincosf(0.5f*th, &st, &ct);
        float spa, cpa; sincosf(0.5f*(phi+om), &spa, &cpa); // ep = (cpa, -spa)
        float sma, cma; sincosf(0.5f*(phi-om), &sma, &cma); // em = (cma, -sma)
        gbuf[t][0] =  cpa*ct;  gbuf[t][1] = -spa*ct;   // R00 = ep*c
        gbuf[t][2] = -cma*st;  gbuf[t][3] = -sma*st;   // R01 = -conj(em)*s
        gbuf[t][4] =  cma*st;  gbuf[t][5] = -sma*st;   // R10 = em*s
        gbuf[t][6] =  cpa*ct;  gbuf[t][7] =  spa*ct;   // R11 = conj(ep)*c
    }
    __syncthreads();

    // ---- Layer 1 via WMMA f32 16x16x4: two 16-sample tiles per wave ----
    // B1[K][N] = [W1[N][0]; W1[N][1]; b1[N]; 0] for N<4 else 0   (K = r + 2*hi)
    v2f b1v;
    if (N < 4) {
        if (hi == 0) { b1v[0] = w1[N*2+0]; b1v[1] = w1[N*2+1]; }
        else         { b1v[0] = b1[N];     b1v[1] = 0.0f; }
    } else { b1v[0] = 0.0f; b1v[1] = 0.0f; }

    #pragma unroll
    for (int T = 0; T < 2; ++T) {
        const int srow = wave*32 + T*16 + N;     // this lane's A-row sample
        // A1[M][K] = [x0, x1, 1, 0]  (bias folded at K=2)
        v2f a;
        if (hi == 0) { a[0] = xs[2*srow]; a[1] = xs[2*srow+1]; }  // K=0,1
        else         { a[0] = 1.0f;       a[1] = 0.0f; }          // K=2,3
        v8f c = {0.f,0.f,0.f,0.f,0.f,0.f,0.f,0.f};
        v8f d = __builtin_amdgcn_wmma_f32_16x16x4_f32(false, a, false, b1v, (short)0, c, false, false);
        if (N < 4) {                              // store RAW pre-activations
            #pragma unroll
            for (int r = 0; r < 8; ++r)
                h1s[(wave*32 + T*16 + r + 8*hi)*4 + N] = d[r];
        }
    }
    __syncthreads();

    // Per-sample tanh on own 4 values (vectorized LDS round-trip)
    {
        float4* h1v4 = (float4*)h1s;
        float4 v = h1v4[t];
        v.x = tanh_fast(v.x); v.y = tanh_fast(v.y);
        v.z = tanh_fast(v.z); v.w = tanh_fast(v.w);
        h1v4[t] = v;
    }
    __syncthreads();

    // ---- Layer 2 via WMMA: A2 = tanh(h1), bias via C operand ----
    v2f b2v;
    if (N < 2) { b2v[0] = w2[N*4 + 2*hi + 0]; b2v[1] = w2[N*4 + 2*hi + 1]; }
    else       { b2v[0] = 0.f; b2v[1] = 0.f; }
    const float cb = (N < 2) ? b2[N] : 0.f;

    #pragma unroll
    for (int T = 0; T < 2; ++T) {
        const int srow = wave*32 + T*16 + N;
        v2f a2;
        a2[0] = h1s[srow*4 + 2*hi + 0];
        a2[1] = h1s[srow*4 + 2*hi + 1];
        v8f cc = {cb,cb,cb,cb,cb,cb,cb,cb};
        v8f d2 = __builtin_amdgcn_wmma_f32_16x16x4_f32(false, a2, false, b2v, (short)0, cc, false, false);
        if (N < 2) {                              // store RAW pre-activations
            #pragma unroll
            for (int r = 0; r < 8; ++r)
                h2s[(wave*32 + T*16 + r + 8*hi)*2 + N] = d2[r];
        }
    }
    __syncthreads();

    // ---- Per-thread 2-qubit statevector simulation (state[q0][q1], 4 complex) ----
    const float h0  = tanh_fast(h2s[2*t]);
    const float h1q = tanh_fast(h2s[2*t+1]);
    float s0, c0, s1, c1;
    __sincosf(0.5f*h0,  &s0, &c0);   // |arg| <= 0.5: hardware sin/cos is accurate here
    __sincosf(0.5f*h1q, &s1, &c1);

    cplx a00, a01, a10, a11;
    // Peeled layer-0 embedding applied to |00>:  m = (RX0 x RX1)|00>
    cplx m00 = { c0*c1, 0.f };
    cplx m01 = { 0.f, -c0*s1 };
    cplx m10 = { 0.f, -s0*c1 };
    cplx m11 = { -s0*s1, 0.f };
    #pragma unroll
    for (int l = 0; l < QLAYERS; ++l) {
        if (l > 0) {
            // RX(h0) on qubit0:  n[i][k] = c*a[i][k] - i*s*a[1-i][k]
            cplx n00 = { fmaf(c0,a00.x,  s0*a10.y), fmaf(c0,a00.y, -s0*a10.x) };
            cplx n01 = { fmaf(c0,a01.x,  s0*a11.y), fmaf(c0,a01.y, -s0*a11.x) };
            cplx n10 = { fmaf(c0,a10.x,  s0*a00.y), fmaf(c0,a10.y, -s0*a00.x) };
            cplx n11 = { fmaf(c0,a11.x,  s0*a01.y), fmaf(c0,a11.y, -s0*a01.x) };
            // RX(h1) on qubit1
            m00 = { fmaf(c1,n00.x,  s1*n01.y), fmaf(c1,n00.y, -s1*n01.x) };
            m01 = { fmaf(c1,n01.x,  s1*n00.y), fmaf(c1,n01.y, -s1*n00.x) };
            m10 = { fmaf(c1,n10.x,  s1*n11.y), fmaf(c1,n10.y, -s1*n11.x) };
            m11 = { fmaf(c1,n11.x,  s1*n10.y), fmaf(c1,n11.y, -s1*n10.x) };
        }
        // Rot on wire0 (first index)
        const float* gA = gbuf[2*l+0];
        cplx r00={gA[0],gA[1]}, r01={gA[2],gA[3]}, r10={gA[4],gA[5]}, r11={gA[6],gA[7]};
        cplx p00 = cmac(cmul(r00,m00), r01, m10);
        cplx p01 = cmac(cmul(r00,m01), r01, m11);
        cplx p10 = cmac(cmul(r10,m00), r11, m10);
        cplx p11 = cmac(cmul(r10,m01), r11, m11);
        // Rot on wire1 (second index): q[i][j'] = sum_j R1[j'][j] p[i][j]
        const float* gB = gbuf[2*l+1];
        cplx u00={gB[0],gB[1]}, u01={gB[2],gB[3]}, u10={gB[4],gB[5]}, u11={gB[6],gB[7]};
        cplx q00 = cmac(cmul(u00,p00), u01, p01);
        cplx q01 = cmac(cmul(u10,p00), u11, p01);
        cplx q10 = cmac(cmul(u00,p10), u01, p11);
        cplx q11 = cmac(cmul(u10,p10), u11, p11);
        // CNOT(0,1): swap (1,0)<->(1,1);  CNOT(1,0): swap (0,1)<->(1,1)
        cplx tsw = q10; q10 = q11; q11 = tsw;
        tsw = q01; q01 = q11; q11 = tsw;
        a00 = q00; a01 = q01; a10 = q10; a11 = q11;
    }
    const float p00 = a00.x*a00.x + a00.y*a00.y;
    const float p01 = a01.x*a01.x + a01.y*a01.y;
    const float p10 = a10.x*a10.x + a10.y*a10.y;
    const float p11 = a11.x*a11.x + a11.y*a11.y;
    const float z0 = (p00 + p01) - (p10 + p11);
    const float z1 = (p00 + p10) - (p01 + p11);
    const float q0v = (z0 + 1.f) * 0.5f;
    const float q1v = (z1 + 1.f) * 0.5f;
    const float lin = fmaf(q0v, pw[0], fmaf(q1v, pw[1], pb[0]));
    const float y = __fdividef(1.f, 1.f + __expf(-lin));
    if (g0 < B) out[g0] = y;
}

extern "C" void kernel_launch(void* const* d_in, const int* in_sizes, int n_in,
                              void* d_out, int out_size, void* d_ws, size_t ws_size,
                              hipStream_t stream) {
    const float* x  = (const float*)d_in[0];
    const float* w1 = (const float*)d_in[1];
    const float* b1 = (const float*)d_in[2];
    const float* w2 = (const float*)d_in[3];
    const float* b2 = (const float*)d_in[4];
    const float* qw = (const float*)d_in[5];
    const float* pw = (const float*)d_in[6];
    const float* pb = (const float*)d_in[7];
    float* out = (float*)d_out;
    const int B = in_sizes[0] / 2;
    const int grid = (B + BLOCK - 1) / BLOCK;
    hybrid_qnn_kernel<<<grid, BLOCK, 0, stream>>>(x, w1, b1, w2, b2, qw, pw, pb, out, B);
}